// Expert_24696061952302
// MI455X (gfx1250) — compile-verified
//
#include <hip/hip_runtime.h>

typedef float v2f __attribute__((ext_vector_type(2)));
typedef float v8f __attribute__((ext_vector_type(8)));

#define IN_DIM 128
#define HID 128
#define OUT 64

// ---------------------------------------------------------------------------
// deg[i] = 1 (self loop)
__global__ __launch_bounds__(256) void k_init_deg(float* __restrict__ deg, int n) {
    int i = blockIdx.x * blockDim.x + threadIdx.x;
    if (i < n) deg[i] = 1.0f;
}

// deg[dst[e]] += 1
__global__ __launch_bounds__(256) void k_edge_deg(const int* __restrict__ dst, int E,
                                                  float* __restrict__ deg) {
    int e = blockIdx.x * blockDim.x + threadIdx.x;
    if (e < E) atomicAdd(&deg[dst[e]], 1.0f);
}

// dinv[i] = rsqrt(deg[i])
__global__ __launch_bounds__(256) void k_rsqrt(const float* __restrict__ deg,
                                               float* __restrict__ dinv, int n) {
    int i = blockIdx.x * blockDim.x + threadIdx.x;
    if (i < n) dinv[i] = rsqrtf(deg[i]);
}

// ---------------------------------------------------------------------------
// H = X (nrows x 128) @ W (128 x NTILES*16), fp32 WMMA 16x16x4.
// One wave32 per 16-row tile; NTILES accumulator tiles held in VGPRs.
template <int NTILES>
__global__ __launch_bounds__(256) void k_gemm_wmma(const float* __restrict__ X,
                                                   const float* __restrict__ W,
                                                   float* __restrict__ H, int nrows) {
    const int lane = threadIdx.x & 31;
    const int wave = threadIdx.x >> 5;
    const int tile = blockIdx.x * (blockDim.x >> 5) + wave;
    if (tile * 16 >= nrows) return;  // whole-wave uniform exit: EXEC stays all-ones

    const int ln = lane & 15;   // 0..15
    const int hi = lane >> 4;   // 0 or 1
    const int ldw = NTILES * 16;

    v8f acc[NTILES] = {};

    const float* xrow = X + (size_t)(tile * 16 + ln) * IN_DIM;

    for (int k = 0; k < IN_DIM; k += 4) {
        // A fragment (16x4 f32): lanes 0-15 -> K=k,k+1 ; lanes 16-31 -> K=k+2,k+3
        v2f a;
        a.x = xrow[k + 2 * hi + 0];
        a.y = xrow[k + 2 * hi + 1];

        const float* wr = W + (size_t)(k + 2 * hi) * ldw;
#pragma unroll
        for (int n = 0; n < NTILES; ++n) {
            // B fragment (4x16 f32): lanes 0-15 -> K=k,k+1 ; lanes 16-31 -> K=k+2,k+3
            v2f b;
            b.x = wr[n * 16 + ln];
            b.y = wr[ldw + n * 16 + ln];
            acc[n] = __builtin_amdgcn_wmma_f32_16x16x4_f32(
                false, a, false, b, (short)0, acc[n], false, false);
        }
    }

    // C/D layout: VGPR r, lanes 0-15 -> M=r, N=ln ; lanes 16-31 -> M=r+8, N=ln
#pragma unroll
    for (int n = 0; n < NTILES; ++n) {
#pragma unroll
        for (int r = 0; r < 8; ++r) {
            int orow = tile * 16 + r + 8 * hi;
            H[(size_t)orow * ldw + n * 16 + ln] = acc[n][r];
        }
    }
}

// ---------------------------------------------------------------------------
// A[i][c] = H[i][c]*dinv[i]^2 + bias[c]  (self-loop message + bias init)
__global__ __launch_bounds__(256) void k_self_bias(const float* __restrict__ H,
                                                   const float* __restrict__ dinv,
                                                   const float* __restrict__ bias,
                                                   float* __restrict__ A,
                                                   int n, int dim) {
    size_t idx = (size_t)blockIdx.x * blockDim.x + threadIdx.x;
    size_t total = (size_t)n * dim;
    if (idx < total) {
        int i = (int)(idx / dim);
        int c = (int)(idx % dim);
        float s = dinv[i];
        A[idx] = H[idx] * (s * s) + bias[c];
    }
}

// One wave32 per edge: A[dst] += H[src] * dinv[src]*dinv[dst]
template <int DIM>
__global__ __launch_bounds__(256) void k_scatter(const float* __restrict__ H,
                                                 const int* __restrict__ src,
                                                 const int* __restrict__ dst,
                                                 const float* __restrict__ dinv,
                                                 float* __restrict__ A, int E) {
    const int lane = threadIdx.x & 31;
    const int e = (int)(((size_t)blockIdx.x * blockDim.x + threadIdx.x) >> 5);
    if (e >= E) return;
    const int s = src[e];
    const int d = dst[e];
    const float norm = dinv[s] * dinv[d];
    constexpr int VPT = DIM / 32;  // 4 (dim=128) or 2 (dim=64)
    const float* hp = H + (size_t)s * DIM + lane * VPT;
    float* ap = A + (size_t)d * DIM + lane * VPT;
#pragma unroll
    for (int j = 0; j < VPT; ++j) atomicAdd(&ap[j], hp[j] * norm);
}

// in-place ReLU
__global__ __launch_bounds__(256) void k_relu(float* __restrict__ A, size_t total) {
    size_t idx = (size_t)blockIdx.x * blockDim.x + threadIdx.x;
    if (idx < total) A[idx] = fmaxf(A[idx], 0.0f);
}

// ---------------------------------------------------------------------------
extern "C" void kernel_launch(void* const* d_in, const int* in_sizes, int n_in,
                              void* d_out, int out_size, void* d_ws, size_t ws_size,
                              hipStream_t stream) {
    const float* x   = (const float*)d_in[0];
    const int*   ei  = (const int*)d_in[1];   // edge_index, shape (2, E) row-major
    const float* W1  = (const float*)d_in[2];
    const float* b1  = (const float*)d_in[3];
    const float* W2  = (const float*)d_in[4];
    const float* b2  = (const float*)d_in[5];
    float*       out = (float*)d_out;

    const int n = in_sizes[0] / IN_DIM;   // 100000
    const int E = in_sizes[1] / 2;        // 1600000
    const int* srcIdx = ei;
    const int* dstIdx = ei + E;

    float* ws   = (float*)d_ws;
    float* deg  = ws;                       // n
    float* dinv = deg + n;                  // n
    float* h    = dinv + n;                 // n*128 (reused as n*64 for layer 2)
    float* agg  = h + (size_t)n * HID;      // n*128

    // --- normalization ---
    k_init_deg<<<(n + 255) / 256, 256, 0, stream>>>(deg, n);
    k_edge_deg<<<(E + 255) / 256, 256, 0, stream>>>(dstIdx, E, deg);
    k_rsqrt<<<(n + 255) / 256, 256, 0, stream>>>(deg, dinv, n);

    const int tiles = (n + 15) / 16;                 // 6250 (exact)
    const int gemmBlocks = (tiles + 7) / 8;          // 8 waves per 256-thread block
    const int scatterBlocks = (int)(((size_t)E * 32 + 255) / 256);

    // --- layer 1 ---
    k_gemm_wmma<8><<<gemmBlocks, 256, 0, stream>>>(x, W1, h, n);
    {
        size_t total = (size_t)n * HID;
        k_self_bias<<<(int)((total + 255) / 256), 256, 0, stream>>>(h, dinv, b1, agg, n, HID);
        k_scatter<HID><<<scatterBlocks, 256, 0, stream>>>(h, srcIdx, dstIdx, dinv, agg, E);
        k_relu<<<(int)((total + 255) / 256), 256, 0, stream>>>(agg, total);
    }

    // --- layer 2 ---
    k_gemm_wmma<4><<<gemmBlocks, 256, 0, stream>>>(agg, W2, h, n);
    {
        size_t total = (size_t)n * OUT;
        k_self_bias<<<(int)((total + 255) / 256), 256, 0, stream>>>(h, dinv, b2, out, n, OUT);
        k_scatter<OUT><<<scatterBlocks, 256, 0, stream>>>(h, srcIdx, dstIdx, dinv, out, E);
    }
}